// Net_29291676958726
// MI455X (gfx1250) — compile-verified
//
#include <hip/hip_runtime.h>
#include <cstdint>

// Patch-based denoiser: per-pixel 5x5 variance (ddof=1) + median + relu blend.
// CDNA5 strategy: async global->LDS DMA (ASYNCcnt) stages the 36x36 halo tile;
// zero padding handled by per-lane EXEC predication of the async op / LDS store.
// Median-of-25 via provably-correct forgetful selection with pairwise minmax
// (129 compare-exchanges + one v_med3_f32). Pure order statistics -> no WMMA.

#define IMG_H 512
#define IMG_W 512
#define BATCH 16
#define TILE  32
#define HALO  2
#define TW    (TILE + 2 * HALO)   // 36

// compare-exchange: p = min, q = max (multiset-preserving)
#define S2(p, q) { const float _lo = fminf(p, q); const float _hi = fmaxf(p, q); (p) = _lo; (q) = _hi; }

__global__ __launch_bounds__(256)
void Net_29291676958726_kernel(const float* __restrict__ x,
                               const float* __restrict__ noise_var,
                               const float* __restrict__ noise_bias,
                               float* __restrict__ out)
{
    __shared__ float tile[TW * TW];

    const int bx = blockIdx.x, by = blockIdx.y, b = blockIdx.z;
    const int tx = threadIdx.x, ty = threadIdx.y;       // 32 x 8
    const int tid = ty * 32 + tx;

    const int x0 = bx * TILE - HALO;
    const int y0 = by * TILE - HALO;
    const float* img = x + (size_t)b * IMG_H * IMG_W;

    // ---- Stage halo tile into LDS via CDNA5 async global->LDS copies ----
    for (int t = tid; t < TW * TW; t += 256) {
        const int ly = t / TW;
        const int lx = t - ly * TW;
        const int iy = y0 + ly;
        const int ix = x0 + lx;
        // Low 32 bits of the flat address of a __shared__ element == LDS byte offset.
        const uint32_t laddr = (uint32_t)(uintptr_t)&tile[t];
        if (iy >= 0 && iy < IMG_H && ix >= 0 && ix < IMG_W) {
            const uint64_t gaddr =
                (uint64_t)(uintptr_t)(img + (size_t)iy * IMG_W + ix);
            asm volatile("global_load_async_to_lds_b32 %0, %1, off"
                         :: "v"(laddr), "v"(gaddr) : "memory");
        } else {
            tile[t] = 0.0f;   // zero padding, matches jnp.pad
        }
    }
    // Wave-local fence on the async DMA counter, then workgroup barrier.
    asm volatile("s_wait_asynccnt 0x0" ::: "memory");
    __syncthreads();

    const float nv = noise_var[0];
    const float nb = noise_bias[0];

    #pragma unroll
    for (int k = 0; k < 4; ++k) {
        const int lyy = ty + k * 8;            // output row 0..31 within tile
        const int gx = bx * TILE + tx;
        const int gy = by * TILE + lyy;

        float v[25];
        float sum = 0.0f, sumsq = 0.0f;
        #pragma unroll
        for (int i = 0; i < 5; ++i) {
            #pragma unroll
            for (int j = 0; j < 5; ++j) {
                const float val = tile[(lyy + i) * TW + (tx + j)];
                v[i * 5 + j] = val;
                sum += val;
                sumsq = fmaf(val, val, sumsq);
            }
        }
        const float xc = v[12];                // center pixel
        // unbiased variance: (sum x^2 - (sum x)^2 / N) / (N - 1)
        const float var = (sumsq - sum * sum * (1.0f / 25.0f)) * (1.0f / 24.0f);

        // ---- median of 25: forgetful selection, window = 14 ----
        // A discarded window-min is <= 13 elements (rank <= 12); a discarded
        // window-max is >= 13 elements (rank >= 14); neither can be the
        // 13th-smallest. After 11 steps, median = med3 of the 3 survivors.
        float a[14];
        #pragma unroll
        for (int i = 0; i < 14; ++i) a[i] = v[i];

        #pragma unroll
        for (int s = 0; s < 11; ++s) {
            const int m = 14 - s;              // active window a[0..m-1]
            // Phase 1: pair ends -> global min in front(+mid), global max in back(+mid)
            #pragma unroll
            for (int i = 0; i < m / 2; ++i) { S2(a[i], a[m - 1 - i]); }
            // Phase 2: min -> a[0]   (touches only front half + middle)
            #pragma unroll
            for (int i = 1; i <= (m - 1) / 2; ++i) { S2(a[0], a[i]); }
            // Phase 3: max -> a[m-1] (touches only back half + middle)
            #pragma unroll
            for (int i = m / 2; i < m - 1; ++i) { S2(a[i], a[m - 1]); }
            a[0] = v[14 + s];                  // drop min & max, insert next value
        }
#if __has_builtin(__builtin_amdgcn_fmed3f)
        const float med = __builtin_amdgcn_fmed3f(a[0], a[1], a[2]);  // v_med3_f32
#else
        const float med = fmaxf(fminf(fmaxf(a[0], a[1]), a[2]), fminf(a[0], a[1]));
#endif

        const float y = xc - nv / (var + 1e-10f) * (xc - med + nb);
        out[((size_t)b * IMG_H + gy) * IMG_W + gx] = fmaxf(y, 0.0f);
    }
}

extern "C" void kernel_launch(void* const* d_in, const int* in_sizes, int n_in,
                              void* d_out, int out_size, void* d_ws, size_t ws_size,
                              hipStream_t stream) {
    (void)in_sizes; (void)n_in; (void)d_ws; (void)ws_size; (void)out_size;
    const float* x          = (const float*)d_in[0];
    const float* noise_var  = (const float*)d_in[1];
    const float* noise_bias = (const float*)d_in[2];
    float* out              = (float*)d_out;

    dim3 grid(IMG_W / TILE, IMG_H / TILE, BATCH);   // 16 x 16 x 16
    dim3 block(32, 8);                              // 8 wave32s
    Net_29291676958726_kernel<<<grid, block, 0, stream>>>(x, noise_var, noise_bias, out);
}